// ObjEncoder_1589137899524
// MI455X (gfx1250) — compile-verified
//
#include <hip/hip_runtime.h>
#include <stdint.h>

typedef __attribute__((ext_vector_type(2))) float v2f;
typedef __attribute__((ext_vector_type(8))) float v8f;

#define WAVE 32

// ---------------------------------------------------------------------------
// WMMA helper: D = A(16x4,f32) * B(4x16,f32) + C, f32 accumulate.
// Fragment layouts per CDNA5 ISA 7.12.2 (32-bit A 16x4 / B 4x16 / C,D 16x16).
// ---------------------------------------------------------------------------
__device__ __forceinline__ v8f wmma4(v2f a, v2f b, v8f c) {
  return __builtin_amdgcn_wmma_f32_16x16x4_f32(
      /*neg_a=*/false, a, /*neg_b=*/false, b,
      /*c_mod=*/(short)0, c, /*reuse_a=*/false, /*reuse_b=*/false);
}

// ---------------------------------------------------------------------------
// Generic LDS->LDS (or LDS->global) tile GEMM: Y[R,COUT] = act(X[R,K] @ Wt + b)
// X row-major with pitch PIN in LDS; Wt is (K x COUT) row-major in global
// (pre-transposed, zero K-padded); Y row-major with pitch POUT.
// Register-blocks MB M-tiles per task so each B fragment feeds MB WMMAs.
// ---------------------------------------------------------------------------
template<int ROWS, int K, int COUT, int PIN, int POUT, bool RELU, int MB>
__device__ __forceinline__ void gemm_lds(const float* Xs,
                                         const float* __restrict__ Wt,
                                         const float* __restrict__ bias,
                                         float* Ys, int lane, int wid, int nw) {
  constexpr int MT = ROWS / 16, NT = COUT / 16;
  static_assert(MT % MB == 0, "MB must divide MT");
  constexpr int MG = MT / MB;
  const int ml = lane & 15, hi = lane >> 4;
  for (int t = wid; t < MG * NT; t += nw) {
    const int mg = t / NT, nt = t % NT;
    v8f acc[MB];
#pragma unroll
    for (int mi = 0; mi < MB; ++mi) acc[mi] = {};
    const float* xr = Xs + (size_t)(mg * MB * 16 + ml) * PIN;
    const float* wc = Wt + nt * 16 + ml;
    for (int k = 0; k < K; k += 4) {
      v2f b;
      b[0] = wc[(size_t)(k + 2 * hi + 0) * COUT];
      b[1] = wc[(size_t)(k + 2 * hi + 1) * COUT];
#pragma unroll
      for (int mi = 0; mi < MB; ++mi) {
        v2f a;
        a[0] = xr[(size_t)mi * 16 * PIN + k + 2 * hi + 0];
        a[1] = xr[(size_t)mi * 16 * PIN + k + 2 * hi + 1];
        acc[mi] = wmma4(a, b, acc[mi]);
      }
    }
    const float bb = bias[nt * 16 + ml];
    float* yb = Ys + nt * 16 + ml;
#pragma unroll
    for (int mi = 0; mi < MB; ++mi) {
#pragma unroll
      for (int v = 0; v < 8; ++v) {
        float y = acc[mi][v] + bb;
        if (RELU) y = fmaxf(y, 0.0f);
        yb[(size_t)((mg * MB + mi) * 16 + v + 8 * hi) * POUT] = y;
      }
    }
  }
}

// ---------------------------------------------------------------------------
// Final MLP layer fused with ReLU + max over NSAMP rows per center.
// out[ci*outStride + ch] = max_over_samples relu(X @ Wt + b).
// All NSAMP/16 sample-tiles of a center share each B fragment.
// ---------------------------------------------------------------------------
template<int NSAMP, int K, int COUT, int PIN, int CHUNKN>
__device__ __forceinline__ void gemm_relu_max(const float* Xs,
                                              const float* __restrict__ Wt,
                                              const float* __restrict__ bias,
                                              float* __restrict__ out, int outStride,
                                              int lane, int wid, int nw) {
  constexpr int MTC = NSAMP / 16, NT = COUT / 16;
  const int ml = lane & 15, hi = lane >> 4;
  for (int t = wid; t < CHUNKN * NT; t += nw) {
    const int ci = t / NT, nt = t % NT;
    v8f acc[MTC];
#pragma unroll
    for (int mt = 0; mt < MTC; ++mt) acc[mt] = {};
    const float* xr = Xs + (size_t)(ci * NSAMP + ml) * PIN;
    const float* wc = Wt + nt * 16 + ml;
    for (int k = 0; k < K; k += 4) {
      v2f b;
      b[0] = wc[(size_t)(k + 2 * hi + 0) * COUT];
      b[1] = wc[(size_t)(k + 2 * hi + 1) * COUT];
#pragma unroll
      for (int mt = 0; mt < MTC; ++mt) {
        v2f a;
        a[0] = xr[(size_t)mt * 16 * PIN + k + 2 * hi + 0];
        a[1] = xr[(size_t)mt * 16 * PIN + k + 2 * hi + 1];
        acc[mt] = wmma4(a, b, acc[mt]);
      }
    }
    const float bb = bias[nt * 16 + ml];
    float cmax = 0.0f;  // relu output is >= 0
#pragma unroll
    for (int mt = 0; mt < MTC; ++mt)
#pragma unroll
      for (int v = 0; v < 8; ++v) cmax = fmaxf(cmax, fmaxf(acc[mt][v] + bb, 0.0f));
    // fold the two row-halves (lanes L and L^16 hold rows r and r+8)
    cmax = fmaxf(cmax, __shfl_xor(cmax, 16, WAVE));
    if (hi == 0) out[(size_t)ci * outStride + nt * 16 + ml] = cmax;
  }
}

// ---------------------------------------------------------------------------
// Weight prep: Wt[k][n] = (k < cin) ? W[n][k] : 0   (K padded to mult of 4)
// ---------------------------------------------------------------------------
__global__ void wt_pad_kernel(const float* __restrict__ W, float* __restrict__ Wt,
                              int cout, int cin, int kp) {
  int i = blockIdx.x * blockDim.x + threadIdx.x;
  if (i >= kp * cout) return;
  int k = i / cout, n = i % cout;
  Wt[i] = (k < cin) ? W[(size_t)n * cin + k] : 0.0f;
}

// ---------------------------------------------------------------------------
// Farthest point sampling; one block per cloud. Writes selected coords.
// Matches reference: first pick is index 0; then argmax of running min-dist.
// ---------------------------------------------------------------------------
__global__ void fps_kernel(const float* __restrict__ pts, int ptStride, int npts,
                           int npoint, float* __restrict__ outXyz) {
  extern __shared__ float smem[];
  float* sx = smem;
  float* sy = sx + npts;
  float* sz = sy + npts;
  float* mind = sz + npts;
  __shared__ float redV[8];
  __shared__ int redI[8];
  __shared__ int sFar;
  const float* base = pts + (size_t)blockIdx.x * npts * ptStride;
  const int tid = threadIdx.x, nthr = blockDim.x;
  const int lane = tid & (WAVE - 1), wid = tid / WAVE, nw = nthr / WAVE;
  for (int p = tid; p < npts; p += nthr) {
    sx[p] = base[(size_t)p * ptStride + 0];
    sy[p] = base[(size_t)p * ptStride + 1];
    sz[p] = base[(size_t)p * ptStride + 2];
    mind[p] = 1e10f;
  }
  if (tid == 0) sFar = 0;
  __syncthreads();
  float* out = outXyz + (size_t)blockIdx.x * npoint * 3;
  for (int it = 0; it < npoint; ++it) {
    const int far = sFar;
    const float cx = sx[far], cy = sy[far], cz = sz[far];
    if (tid == 0) { out[it * 3 + 0] = cx; out[it * 3 + 1] = cy; out[it * 3 + 2] = cz; }
    float bestV = -1.0f;
    int bestI = 0;
    for (int p = tid; p < npts; p += nthr) {
      float dx = sx[p] - cx, dy = sy[p] - cy, dz = sz[p] - cz;
      float m = fminf(mind[p], dx * dx + dy * dy + dz * dz);
      mind[p] = m;
      if (m > bestV) { bestV = m; bestI = p; }
    }
    for (int off = WAVE / 2; off > 0; off >>= 1) {
      float ov = __shfl_down(bestV, off, WAVE);
      int oi = __shfl_down(bestI, off, WAVE);
      if (ov > bestV || (ov == bestV && oi < bestI)) { bestV = ov; bestI = oi; }
    }
    if (lane == 0) { redV[wid] = bestV; redI[wid] = bestI; }
    __syncthreads();
    if (tid == 0) {
      float bv = redV[0]; int bi = redI[0];
      for (int w = 1; w < nw; ++w)
        if (redV[w] > bv || (redV[w] == bv && redI[w] < bi)) { bv = redV[w]; bi = redI[w]; }
      sFar = bi;
    }
    __syncthreads();
  }
}

// ---------------------------------------------------------------------------
// Fused SA stage: ball query + gather + 3-layer MLP (WMMA) + per-center max.
// One block per cloud; all staging in LDS (CDNA5 320KB/WG).
//   G row = [dx,dy,dz, feat[0..CIN-1], 0-pad to KP]
// ---------------------------------------------------------------------------
template<int NPTS, int NCTR, int CIN, int KP, int C1, int C2, int C3, int CHUNK,
         int MB>
__global__ void __launch_bounds__(256) sa_kernel(
    const float* __restrict__ ptsXyz, int xyzStride,
    const float* __restrict__ ptsFeat, int featStride,
    const float* __restrict__ ctrXyz, float r2,
    const float* __restrict__ Wt0, const float* __restrict__ b0,
    const float* __restrict__ Wt1, const float* __restrict__ b1,
    const float* __restrict__ Wt2, const float* __restrict__ b2,
    float* __restrict__ outFeat) {
  constexpr int NS = 32;
  constexpr int ROWS = CHUNK * NS;
  constexpr int PG = KP + 4, P1 = C1 + 4, P2 = C2 + 4;
  extern __shared__ float smem[];
  float* sPx = smem;
  float* sPy = sPx + NPTS;
  float* sPz = sPy + NPTS;
  float* sF = sPz + NPTS;             // NPTS*CIN
  float* sCx = sF + NPTS * CIN;       // NCTR
  float* sCy = sCx + NCTR;
  float* sCz = sCy + NCTR;
  int* sIdx = (int*)(sCz + NCTR);     // NCTR*NS
  float* sG = (float*)(sIdx + NCTR * NS);  // ROWS*PG
  float* sA1 = sG + ROWS * PG;             // ROWS*P1
  float* sA2 = sA1 + ROWS * P1;            // ROWS*P2

  const int tid = threadIdx.x, nthr = blockDim.x;
  const int lane = tid & 31, wid = tid >> 5, nw = nthr >> 5;
  const int cloud = blockIdx.x;
  const float* pxb = ptsXyz + (size_t)cloud * NPTS * xyzStride;
  const float* pfb = ptsFeat + (size_t)cloud * NPTS * featStride;
  const float* cxb = ctrXyz + (size_t)cloud * NCTR * 3;

  for (int p = tid; p < NPTS; p += nthr) {
    sPx[p] = pxb[(size_t)p * xyzStride + 0];
    sPy[p] = pxb[(size_t)p * xyzStride + 1];
    sPz[p] = pxb[(size_t)p * xyzStride + 2];
    for (int c = 0; c < CIN; ++c) sF[p * CIN + c] = pfb[(size_t)p * featStride + c];
  }
  for (int c = tid; c < NCTR; c += nthr) {
    sCx[c] = cxb[c * 3 + 0]; sCy[c] = cxb[c * 3 + 1]; sCz[c] = cxb[c * 3 + 2];
  }
  __syncthreads();

  // Ball query: first NS in-range indices ascending, pad with first index.
  for (int c = wid; c < NCTR; c += nw) {
    const float cx = sCx[c], cy = sCy[c], cz = sCz[c];
    int cnt = 0, firstI = 0;
    bool haveFirst = false;
    for (int base = 0; base < NPTS; base += WAVE) {
      const int p = base + lane;
      bool inr = false;
      if (p < NPTS) {
        float dx = sPx[p] - cx, dy = sPy[p] - cy, dz = sPz[p] - cz;
        inr = (dx * dx + dy * dy + dz * dz) < r2;
      }
      unsigned msk = (unsigned)__ballot(inr);
      if (!haveFirst && msk) { firstI = base + __ffs(msk) - 1; haveFirst = true; }
      int pos = cnt + __popc(msk & ((1u << lane) - 1u));
      if (inr && pos < NS) sIdx[c * NS + pos] = p;
      cnt += __popc(msk);
      if (cnt >= NS) break;
    }
    if (cnt < NS)
      for (int j = cnt + lane; j < NS; j += WAVE) sIdx[c * NS + j] = firstI;
  }
  __syncthreads();

  // Process centers in chunks: gather -> L1 -> L2 -> L3+max
  for (int cb0 = 0; cb0 < NCTR; cb0 += CHUNK) {
    for (int r = tid; r < ROWS; r += nthr) {
      const int c = cb0 + (r >> 5), s = r & 31;
      const int p = sIdx[c * NS + s];
      float* g = sG + (size_t)r * PG;
      g[0] = sPx[p] - sCx[c];
      g[1] = sPy[p] - sCy[c];
      g[2] = sPz[p] - sCz[c];
      for (int k = 0; k < CIN; ++k) g[3 + k] = sF[p * CIN + k];
      for (int k = 3 + CIN; k < KP; ++k) g[k] = 0.0f;
    }
    __syncthreads();
    gemm_lds<ROWS, KP, C1, PG, P1, true, MB>(sG, Wt0, b0, sA1, lane, wid, nw);
    __syncthreads();
    gemm_lds<ROWS, C1, C2, P1, P2, true, MB>(sA1, Wt1, b1, sA2, lane, wid, nw);
    __syncthreads();
    float* outp = outFeat + (size_t)cloud * NCTR * C3 + (size_t)cb0 * C3;
    gemm_relu_max<NS, C2, C3, P2, CHUNK>(sA2, Wt2, b2, outp, C3, lane, wid, nw);
    __syncthreads();
  }
}

// ---------------------------------------------------------------------------
// SA3: per-cloud global MLP over 16 points (concat xyz2+feats2 = 259ch) + max.
// ---------------------------------------------------------------------------
__global__ void __launch_bounds__(256) sa3_kernel(
    const float* __restrict__ xyz2, const float* __restrict__ feats2,
    const float* __restrict__ Wt0, const float* __restrict__ b0,
    const float* __restrict__ Wt1, const float* __restrict__ b1,
    const float* __restrict__ Wt2, const float* __restrict__ b2,
    float* __restrict__ out) {
  constexpr int R = 16, KP = 260, C1 = 256, C2 = 512, C3 = 768;
  constexpr int PX = KP + 4, P1 = C1 + 4, P2 = C2 + 4;
  extern __shared__ float smem[];
  float* sX = smem;
  float* sA1 = sX + R * PX;
  float* sA2 = sA1 + R * P1;
  const int tid = threadIdx.x, nthr = blockDim.x;
  const int lane = tid & 31, wid = tid >> 5, nw = nthr >> 5;
  const int cloud = blockIdx.x;
  const float* xz = xyz2 + (size_t)cloud * R * 3;
  const float* ft = feats2 + (size_t)cloud * R * 256;
  for (int i = tid; i < R * KP; i += nthr) {
    const int r = i / KP, k = i % KP;
    float v;
    if (k < 3) v = xz[r * 3 + k];
    else if (k < 259) v = ft[(size_t)r * 256 + (k - 3)];
    else v = 0.0f;
    sX[r * PX + k] = v;
  }
  __syncthreads();
  gemm_lds<R, KP, C1, PX, P1, true, 1>(sX, Wt0, b0, sA1, lane, wid, nw);
  __syncthreads();
  gemm_lds<R, C1, C2, P1, P2, true, 1>(sA1, Wt1, b1, sA2, lane, wid, nw);
  __syncthreads();
  gemm_relu_max<16, C2, C3, P2, 1>(sA2, Wt2, b2, out + (size_t)cloud * C3, C3,
                                   lane, wid, nw);
}

// ---------------------------------------------------------------------------
// Final FC (no relu): Y(2048x768) = X @ Wt + b ; one 16-row band per block.
// ---------------------------------------------------------------------------
__global__ void __launch_bounds__(256) fc_kernel(const float* __restrict__ X,
                                                 const float* __restrict__ Wt,
                                                 const float* __restrict__ b,
                                                 float* __restrict__ Y) {
  constexpr int K = 768, COUT = 768, P = K + 4;
  extern __shared__ float smem[];
  float* sX = smem;  // 16*P
  const int tid = threadIdx.x, nthr = blockDim.x;
  const int lane = tid & 31, wid = tid >> 5, nw = nthr >> 5;
  const float* xb = X + (size_t)blockIdx.x * 16 * K;
  for (int i = tid; i < 16 * K; i += nthr) {
    const int r = i / K, k = i % K;
    sX[r * P + k] = xb[i];
  }
  __syncthreads();
  gemm_lds<16, K, COUT, P, COUT, false, 1>(sX, Wt, b,
                                           Y + (size_t)blockIdx.x * 16 * COUT,
                                           lane, wid, nw);
}

// ---------------------------------------------------------------------------
// Host launcher
// ---------------------------------------------------------------------------
extern "C" void kernel_launch(void* const* d_in, const int* in_sizes, int n_in,
                              void* d_out, int out_size, void* d_ws, size_t ws_size,
                              hipStream_t stream) {
  (void)in_sizes; (void)n_in; (void)out_size; (void)ws_size;
  const int M = 32 * 64;  // 2048 clouds

  const float* obj = (const float*)d_in[0];
  const float* W[10]; const float* B[10];
  for (int l = 0; l < 10; ++l) {
    W[l] = (const float*)d_in[1 + 2 * l];
    B[l] = (const float*)d_in[2 + 2 * l];
  }
  // (cout, cin, kp) for sa1_0..2, sa2_0..2, sa3_0..2, fc
  const int COUTv[10] = {64, 64, 128, 128, 128, 256, 256, 512, 768, 768};
  const int CINv[10]  = {6, 64, 64, 131, 128, 128, 259, 256, 512, 768};
  const int KPv[10]   = {8, 64, 64, 132, 128, 128, 260, 256, 512, 768};

  float* base = (float*)d_ws;
  size_t off = 0;
  auto take = [&](size_t n) { float* p = base + off; off += (n + 63) & ~(size_t)63; return p; };
  float* Wt[10];
  for (int l = 0; l < 10; ++l) Wt[l] = take((size_t)KPv[l] * COUTv[l]);
  float* xyz1   = take((size_t)M * 32 * 3);
  float* feats1 = take((size_t)M * 32 * 128);
  float* xyz2   = take((size_t)M * 16 * 3);
  float* feats2 = take((size_t)M * 16 * 256);
  float* feats3 = take((size_t)M * 768);

  // 1) weight transpose + K-pad
  for (int l = 0; l < 10; ++l) {
    int total = KPv[l] * COUTv[l];
    wt_pad_kernel<<<(total + 255) / 256, 256, 0, stream>>>(W[l], Wt[l], COUTv[l], CINv[l], KPv[l]);
  }

  // 2) SA1: FPS(1024->32) then fused ball-query(0.2)+MLP(6->64->64->128)+max
  fps_kernel<<<M, 256, 4 * 1024 * sizeof(float), stream>>>(obj, 6, 1024, 32, xyz1);
  {
    const int smem1 = (3 * 1024 + 1024 * 3 + 3 * 32) * 4 + 32 * 32 * 4 +
                      (256 * 12 + 256 * 68 + 256 * 68) * 4;  // 180,608 B
    auto kfn = sa_kernel<1024, 32, 3, 8, 64, 64, 128, 8, 4>;
    hipFuncSetAttribute((const void*)kfn, hipFuncAttributeMaxDynamicSharedMemorySize, smem1);
    kfn<<<M, 256, smem1, stream>>>(obj, 6, obj + 3, 6, xyz1, 0.04f,
                                   Wt[0], B[0], Wt[1], B[1], Wt[2], B[2], feats1);
  }

  // 3) SA2: FPS(32->16) then fused ball-query(0.4)+MLP(131->128->128->256)+max
  fps_kernel<<<M, 32, 4 * 32 * sizeof(float), stream>>>(xyz1, 3, 32, 16, xyz2);
  {
    const int smem2 = (3 * 32 + 32 * 128 + 3 * 16) * 4 + 16 * 32 * 4 +
                      (128 * 136 + 128 * 132 + 128 * 132) * 4;  // 223,808 B
    auto kfn = sa_kernel<32, 16, 128, 132, 128, 128, 256, 4, 4>;
    hipFuncSetAttribute((const void*)kfn, hipFuncAttributeMaxDynamicSharedMemorySize, smem2);
    kfn<<<M, 256, smem2, stream>>>(xyz1, 3, feats1, 128, xyz2, 0.16f,
                                   Wt[3], B[3], Wt[4], B[4], Wt[5], B[5], feats2);
  }

  // 4) SA3: global MLP(259->256->512->768)+max over 16 points
  {
    const int smem3 = (16 * 264 + 16 * 260 + 16 * 516) * 4;  // 66,560 B
    hipFuncSetAttribute((const void*)sa3_kernel, hipFuncAttributeMaxDynamicSharedMemorySize, smem3);
    sa3_kernel<<<M, 256, smem3, stream>>>(xyz2, feats2, Wt[6], B[6], Wt[7], B[7],
                                          Wt[8], B[8], feats3);
  }

  // 5) FC 768->768 (no relu) -> d_out
  fc_kernel<<<M / 16, 256, 16 * 772 * sizeof(float), stream>>>(
      feats3, Wt[9], B[9], (float*)d_out);
}